// EmbeddingLayer_70669391888820
// MI455X (gfx1250) — compile-verified
//
#include <hip/hip_runtime.h>

typedef __attribute__((ext_vector_type(2))) float v2f;
typedef __attribute__((ext_vector_type(8))) float v8f;

#define D_DIM 64
#define B_DIM 2

// Hardware no-return f32 atomic add (CDNA5 GLOBAL_ATOMIC_ADD_F32).
// Forced via inline asm so it can never fall back to a CAS loop.
__device__ __forceinline__ void gadd_f32(float* p, float v) {
  asm volatile("global_atomic_add_f32 %0, %1, off"
               :
               : "v"(p), "v"(v)
               : "memory");
}

// ---------------------------------------------------------------------------
// Kernel 1: zero the nbr accumulator in workspace (float4 stores, coalesced)
// ---------------------------------------------------------------------------
__global__ void zero_ws_kernel(float4* __restrict__ ws, int n4) {
  int i = blockIdx.x * blockDim.x + threadIdx.x;
  if (i < n4) {
    float4 z; z.x = 0.f; z.y = 0.f; z.z = 0.f; z.w = 0.f;
    ws[i] = z;
  }
}

// ---------------------------------------------------------------------------
// Kernel 2: symmetric gated scatter-add over edges.
// One wave (32 lanes) per edge: lane>>4 selects batch b (0/1),
// (lane&15)*4 selects a float4 chunk of the 64-dim row.
// prev (25.6MB) and nbr (25.6MB) are L2-resident on MI455X (192MB L2),
// so the gathers + f32 atomics stay on-chip.
// ---------------------------------------------------------------------------
__global__ void __launch_bounds__(256)
scatter_edges_kernel(const float* __restrict__ prev,
                     const int*   __restrict__ edges,
                     const float* __restrict__ status,
                     float*       __restrict__ nbr,
                     int E, int N) {
  long tid = (long)blockIdx.x * blockDim.x + threadIdx.x;
  long e   = tid >> 5;
  if (e >= E) return;
  int lane = (int)(tid & 31);
  int b    = lane >> 4;          // batch 0 or 1
  int d4   = (lane & 15) * 4;    // dim chunk 0..60

  int   u = edges[2 * e + 0];
  int   v = edges[2 * e + 1];
  float g = status[e];

  const float4 pu = *(const float4*)(prev + ((size_t)b * N + u) * D_DIM + d4);
  const float4 pv = *(const float4*)(prev + ((size_t)b * N + v) * D_DIM + d4);

  float* ov = nbr + ((size_t)b * N + v) * D_DIM + d4;  // receives h_u = pu*g
  float* ou = nbr + ((size_t)b * N + u) * D_DIM + d4;  // receives h_v = pv*g

  gadd_f32(ov + 0, pu.x * g);
  gadd_f32(ov + 1, pu.y * g);
  gadd_f32(ov + 2, pu.z * g);
  gadd_f32(ov + 3, pu.w * g);
  gadd_f32(ou + 0, pv.x * g);
  gadd_f32(ou + 1, pv.y * g);
  gadd_f32(ou + 2, pv.z * g);
  gadd_f32(ou + 3, pv.w * g);
}

// ---------------------------------------------------------------------------
// Kernel 3: fused  out = leaky_relu(nf + ef + nbr @ W^T)
// One wave per 16-row tile of the flattened (B*N, 64) matrix.
// Uses V_WMMA_F32_16X16X4_F32 chained 16x along K=64, for each of the
// 4 column tiles. C is pre-loaded with nf+ef (bias fusion), leaky-relu
// applied to D before the store.
//
// VGPR layouts per CDNA5 ISA 7.12.2:
//  A 16x4 f32 (2 VGPRs):  lane<16 -> K={0,1} of row lane; lane>=16 -> K={2,3}
//  B 4x16 f32 (2 VGPRs):  mirrored; B[k][n] = W[n][k]  (x2 = nbr @ W^T)
//  C/D 16x16 f32 (8 VGPRs): VGPR r = row (r + 8*(lane>>4)), col = lane&15
// ---------------------------------------------------------------------------
__global__ void __launch_bounds__(256)
gemm_bias_lrelu_kernel(const float* __restrict__ nbr,
                       const float* __restrict__ Wm,   // (64,64) row-major W[e][d]
                       const float* __restrict__ nf,
                       const float* __restrict__ ef,
                       float*       __restrict__ out,
                       int ntiles) {
  long tid  = (long)blockIdx.x * blockDim.x + threadIdx.x;
  long tile = tid >> 5;                 // wave-uniform -> EXEC stays all-ones
  if (tile >= ntiles) return;
  int  lane = (int)(tid & 31);
  int  half = lane >> 4;                // 0: K even pair, 1: +2 offset
  int  lm   = lane & 15;
  long row_base = tile * 16;

  // Preload the whole 16x64 A tile for this wave: 16 fragments x v2f.
  v2f afrag[16];
  const float* arow = nbr + (row_base + lm) * D_DIM + 2 * half;
#pragma unroll
  for (int t = 0; t < 16; ++t)
    afrag[t] = *(const v2f*)(arow + 4 * t);

#pragma unroll
  for (int c = 0; c < 4; ++c) {
    int col = c * 16 + lm;

    // Accumulator initialized with node + edge features (C/D layout).
    v8f acc;
#pragma unroll
    for (int r = 0; r < 8; ++r) {
      long rr = row_base + r + 8 * half;
      acc[r] = nf[rr * D_DIM + col] + ef[rr * D_DIM + col];
    }

    // B fragments: b = { W[col][4t+2*half], W[col][4t+2*half+1] }
    const float* wrow = Wm + (size_t)col * D_DIM + 2 * half;
#pragma unroll
    for (int t = 0; t < 16; ++t) {
      v2f bfrag = *(const v2f*)(wrow + 4 * t);
      // 8 args: (neg_a, A, neg_b, B, c_mod, C, reuse_a, reuse_b)
      acc = __builtin_amdgcn_wmma_f32_16x16x4_f32(
          false, afrag[t], false, bfrag, (short)0, acc, false, false);
    }

    // leaky_relu (slope 0.01) + non-temporal streaming store
#pragma unroll
    for (int r = 0; r < 8; ++r) {
      long rr = row_base + r + 8 * half;
      float dv = acc[r];
      dv = dv > 0.f ? dv : 0.01f * dv;
      __builtin_nontemporal_store(dv, out + rr * D_DIM + col);
    }
  }
}

// ---------------------------------------------------------------------------
extern "C" void kernel_launch(void* const* d_in, const int* in_sizes, int n_in,
                              void* d_out, int out_size, void* d_ws, size_t ws_size,
                              hipStream_t stream) {
  const float* prev  = (const float*)d_in[0];  // (B,N,D) f32
  const int*   edges = (const int*)  d_in[1];  // (E,2)   i32
  const float* nf    = (const float*)d_in[2];  // (B,N,D) f32
  const float* ef    = (const float*)d_in[3];  // (B,N,D) f32
  const float* st    = (const float*)d_in[4];  // (E,)    f32
  const float* Wm    = (const float*)d_in[5];  // (D,D)   f32

  float* nbr = (float*)d_ws;                   // (B,N,D) accumulator

  const int BND = in_sizes[0];                 // B*N*D
  const int E   = in_sizes[4];
  const int N   = BND / (B_DIM * D_DIM);

  // 1) zero the accumulator
  int n4 = BND / 4;
  zero_ws_kernel<<<(n4 + 255) / 256, 256, 0, stream>>>((float4*)nbr, n4);

  // 2) edge scatter: one wave per edge, hardware f32 atomics into L2
  long sthreads = (long)E * 32;
  scatter_edges_kernel<<<(int)((sthreads + 255) / 256), 256, 0, stream>>>(
      prev, edges, st, nbr, E, N);

  // 3) fused WMMA GEMM + bias + leaky_relu: one wave per 16-row tile
  int ntiles = (BND / D_DIM) / 16;             // (B*N)/16, exact for these dims
  long gthreads = (long)ntiles * 32;
  gemm_bias_lrelu_kernel<<<(int)((gthreads + 255) / 256), 256, 0, stream>>>(
      nbr, Wm, nf, ef, (float*)d_out, ntiles);
}